// TypeSpecificProjector_51324859187281
// MI455X (gfx1250) — compile-verified
//
#include <hip/hip_runtime.h>

#define IN_CH    256
#define HID_CH   256
#define TILE_M   16
#define KPANEL   32
#define XSTRIDE  260   // x tile stride (floats): conflict-free A-fragment reads
#define WSTRIDE2 544   // W panel pair-row stride (floats): 544%64==32 -> conflict-free b64 B reads

typedef float v2f __attribute__((ext_vector_type(2)));
typedef float v8f __attribute__((ext_vector_type(8)));

// ---------------- bucketing kernels ----------------

__global__ void tsp_init_kernel(int* __restrict__ cnt, int* __restrict__ idx, int npad) {
    int i = blockIdx.x * blockDim.x + threadIdx.x;
    if (i < npad) idx[i] = -1;
    if (i < 8)    cnt[i] = 0;
}

__global__ void tsp_count_kernel(const int* __restrict__ nt, int* __restrict__ cnt, int n) {
    int i = blockIdx.x * blockDim.x + threadIdx.x;
    if (i < n) atomicAdd(&cnt[nt[i]], 1);
}

__global__ void tsp_scan_kernel(const int* __restrict__ cnt, int* __restrict__ place, int T) {
    if (blockIdx.x == 0 && threadIdx.x == 0) {
        int run = 0;
        for (int t = 0; t < T; ++t) {
            place[t] = run;
            run += (cnt[t] + TILE_M - 1) & ~(TILE_M - 1);  // pad each segment to 16
        }
    }
}

__global__ void tsp_scatter_kernel(const int* __restrict__ nt, int* __restrict__ place,
                                   int* __restrict__ idx, int n) {
    int i = blockIdx.x * blockDim.x + threadIdx.x;
    if (i < n) {
        int t = nt[i];
        int p = atomicAdd(&place[t], 1);
        idx[p] = i;
    }
}

// ---------------- WMMA GEMM over type-uniform 16-row tiles ----------------

__global__ __launch_bounds__(256) void tsp_gemm_kernel(
    const float* __restrict__ x, const int* __restrict__ nt,
    const float* __restrict__ W, const float* __restrict__ Bv,
    const int* __restrict__ idx, float* __restrict__ out)
{
    __shared__ float s_x[TILE_M * XSTRIDE];
    __shared__ float s_w[(KPANEL / 2) * WSTRIDE2];   // K-paired layout: (k2, n) -> k2*544 + 2n
    __shared__ int   s_idx[TILE_M];
    __shared__ int   s_type;

    const int tid  = threadIdx.x;
    const int base = blockIdx.x * TILE_M;

    if (tid < TILE_M) s_idx[tid] = idx[base + tid];
    if (tid == 0) {
        int i0 = idx[base];                 // valid rows are a prefix of each tile
        s_type = (i0 >= 0) ? nt[i0] : -1;
    }
    __syncthreads();
    const int t = s_type;
    if (t < 0) return;                      // block-uniform: padding-only tile

    // Stage A: 16 rows x 256 cols of x into LDS (float4, coalesced; pad rows -> 0)
    #pragma unroll
    for (int j = 0; j < 4; ++j) {
        int c   = tid + j * 256;            // 1024 float4 chunks
        int row = c >> 6;
        int col = (c & 63) << 2;
        int g   = s_idx[row];
        float4 v = make_float4(0.f, 0.f, 0.f, 0.f);
        if (g >= 0) v = *(const float4*)(x + (size_t)g * IN_CH + col);
        *(float4*)(&s_x[row * XSTRIDE + col]) = v;
    }

    const float* Wt = W + (size_t)t * IN_CH * HID_CH;
    const int lane = tid & 31;
    const int wave = tid >> 5;
    const int g16  = lane >> 4;             // half-wave: selects K pair / M half
    const int n    = lane & 15;             // column within 16-col tile
    const int m    = lane & 15;             // row of A fragment
    const int kh   = g16 * 2;               // K base of this half-wave
    const int ct0  = wave * 2;              // each wave owns 2 of the 16 col tiles
    const int ct1  = ct0 + 1;

    // Per-wave LDS base offsets for B fragments (float units)
    const int bo0 = g16 * WSTRIDE2 + (ct0 * 16 + n) * 2;
    const int bo1 = g16 * WSTRIDE2 + (ct1 * 16 + n) * 2;

    v8f acc0 = {};
    v8f acc1 = {};

    for (int p = 0; p < IN_CH / KPANEL; ++p) {
        __syncthreads();
        const int k0 = p * KPANEL;
        // Stage W panel (KPANEL x 256) into LDS in K-paired layout:
        // s_w[k2*544 + 2n + v] = W[k0 + 2*k2 + v][n]   (v = 0,1)
        #pragma unroll
        for (int j = 0; j < 8; ++j) {
            int c  = tid + j * 256;         // 2048 chunks of 2 columns
            int k2 = c >> 7;                // 16 pair-rows
            int nn = (c & 127) << 1;        // column (even)
            float2 r0 = *(const float2*)(Wt + (size_t)(k0 + 2 * k2)     * HID_CH + nn);
            float2 r1 = *(const float2*)(Wt + (size_t)(k0 + 2 * k2 + 1) * HID_CH + nn);
            *(float4*)(&s_w[k2 * WSTRIDE2 + nn * 2]) = make_float4(r0.x, r1.x, r0.y, r1.y);
        }
        __syncthreads();

        #pragma unroll
        for (int kk = 0; kk < KPANEL; kk += 4) {
            const int k2 = (kk >> 1);
            // A fragment: lane L, VGPR v <-> A[m, 2*(L/16)+v]  (single b64 LDS read)
            v2f a  = *(const v2f*)(&s_x[m * XSTRIDE + k0 + kk + kh]);
            // B fragments: single b64 LDS reads in K-paired layout
            v2f b0 = *(const v2f*)(&s_w[k2 * WSTRIDE2 + bo0]);
            v2f b1 = *(const v2f*)(&s_w[k2 * WSTRIDE2 + bo1]);
            acc0 = __builtin_amdgcn_wmma_f32_16x16x4_f32(
                false, a, false, b0, (short)0, acc0, false, false);
            acc1 = __builtin_amdgcn_wmma_f32_16x16x4_f32(
                false, a, false, b1, (short)0, acc1, false, false);
        }
    }

    // Bias + scattered store. C/D layout: VGPR v <-> M = v + 8*(L/16), N = lane&15.
    const float bias0 = Bv[t * HID_CH + ct0 * 16 + n];
    const float bias1 = Bv[t * HID_CH + ct1 * 16 + n];
    #pragma unroll
    for (int v = 0; v < 8; ++v) {
        int rowm = v + 8 * g16;
        int g = s_idx[rowm];
        if (g >= 0) {
            out[(size_t)g * HID_CH + ct0 * 16 + n] = acc0[v] + bias0;
            out[(size_t)g * HID_CH + ct1 * 16 + n] = acc1[v] + bias1;
        }
    }
}

// ---------------- host entry ----------------

extern "C" void kernel_launch(void* const* d_in, const int* in_sizes, int n_in,
                              void* d_out, int out_size, void* d_ws, size_t ws_size,
                              hipStream_t stream) {
    const float* x  = (const float*)d_in[0];
    const int*   nt = (const int*)  d_in[1];
    const float* W  = (const float*)d_in[2];
    const float* Bv = (const float*)d_in[3];
    float* out = (float*)d_out;

    const int N = in_sizes[1];                  // node_type has N elements
    const int T = in_sizes[3] / HID_CH;         // bias is [T, HID]

    int* cnt   = (int*)d_ws;                    // [8]  per-type counts
    int* place = cnt + 8;                       // [8]  scatter cursors
    int* idx   = cnt + 16;                      // [npad] row permutation (padded -1)

    const int npad  = ((N + TILE_M - 1) / TILE_M + T) * TILE_M;
    const int tiles = npad / TILE_M;
    const int nb    = (N + 255) / 256;

    tsp_init_kernel   <<<(npad + 255) / 256, 256, 0, stream>>>(cnt, idx, npad);
    tsp_count_kernel  <<<nb, 256, 0, stream>>>(nt, cnt, N);
    tsp_scan_kernel   <<<1, 1, 0, stream>>>(cnt, place, T);
    tsp_scatter_kernel<<<nb, 256, 0, stream>>>(nt, place, idx, N);
    tsp_gemm_kernel   <<<tiles, 256, 0, stream>>>(x, nt, W, Bv, idx, out);
}